// UniformLoss_20401094656262
// MI455X (gfx1250) — compile-verified
//
#include <hip/hip_runtime.h>
#include <hip/hip_bf16.h>
#include <math.h>

#define N_IMG   32
#define A_ANCH  16384
#define C_CLS   81
#define TILE    128
#define SCALE_CLAMP 4.135166556742356f   // log(1000/16)

// ---------------------------------------------------------------------------
// Pass A: stage confidence tiles into LDS via the CDNA5 async copy engine,
// compute log-sum-exp, CE, OHEM sort keys, num_pos, and smooth-L1 (positives).
// ---------------------------------------------------------------------------
__global__ __launch_bounds__(TILE)
void pass_a(const float* __restrict__ conf,
            const float* __restrict__ deltas,
            const float* __restrict__ gt,
            const float* __restrict__ anchors,
            const int*   __restrict__ labels,
            float* __restrict__ ce_out,
            unsigned long long* __restrict__ keys,
            int* __restrict__ num_pos,
            float* __restrict__ acc)   // acc[0] = smooth_l1, acc[1] = cls
{
    __shared__ __align__(16) float s_conf[TILE * C_CLS];   // 41472 B per WG

    const int n   = blockIdx.y;
    const int a0  = blockIdx.x * TILE;
    const int tid = threadIdx.x;

    // ---- async global -> LDS copy of the contiguous [TILE x 81] f32 tile ----
    const unsigned long long gaddr =
        (unsigned long long)(conf + ((size_t)n * A_ANCH + a0) * C_CLS); // 16B aligned
    const unsigned int lbase = (unsigned int)(size_t)(&s_conf[0]);
    const int n128 = TILE * C_CLS / 4;   // 2592 x 16B chunks
    for (int i = tid; i < n128; i += TILE) {
        unsigned int loff = lbase + (unsigned int)i * 16u;
        unsigned int voff = (unsigned int)i * 16u;
        asm volatile("global_load_async_to_lds_b128 %0, %1, %2"
                     :: "v"(loff), "v"(voff), "s"(gaddr)
                     : "memory");
    }
    asm volatile("s_wait_asynccnt 0" ::: "memory");   // this wave's DMA done
    __syncthreads();                                  // all waves' LDS visible

    // ---- per-anchor log-sum-exp over 81 classes (stride-81 words: no bank
    //      conflicts in wave32: lane i hits bank (17*i + c) mod 64) ----
    const float* row = &s_conf[tid * C_CLS];
    float m = row[0];
    #pragma unroll 4
    for (int c = 1; c < C_CLS; ++c) m = fmaxf(m, row[c]);
    float ssum = 0.0f;
    #pragma unroll 4
    for (int c = 0; c < C_CLS; ++c) ssum += expf(row[c] - m);
    const float lse = m + logf(ssum);

    const int a       = a0 + tid;
    const size_t idx  = (size_t)n * A_ANCH + a;
    const int lab     = labels[idx];
    const bool pos    = lab > 0;

    ce_out[idx] = lse - row[lab];            // cross-entropy for this anchor

    // OHEM key: descending bg_loss, positives = -inf, stable tie-break by
    // ascending index (larger key == earlier in argsort(-bg_loss)).
    float keyf = pos ? -INFINITY : (lse - row[0]);
    unsigned int bits = __float_as_uint(keyf);
    unsigned int u    = (bits & 0x80000000u) ? ~bits : (bits | 0x80000000u);
    keys[idx] = ((unsigned long long)u << 32) |
                (unsigned long long)(~(unsigned int)a);

    if (pos) {
        atomicAdd(&num_pos[n], 1);
        // smooth L1 on decoded box vs gt
        const float ax1 = anchors[a * 4 + 0], ay1 = anchors[a * 4 + 1];
        const float ax2 = anchors[a * 4 + 2], ay2 = anchors[a * 4 + 3];
        const float w = ax2 - ax1, h = ay2 - ay1;
        const float cx = ax1 + 0.5f * w, cy = ay1 + 0.5f * h;
        const size_t di = idx * 4;
        const float dx = deltas[di + 0], dy = deltas[di + 1];
        const float dw = fminf(deltas[di + 2], SCALE_CLAMP);
        const float dh = fminf(deltas[di + 3], SCALE_CLAMP);
        const float pcx = dx * w + cx, pcy = dy * h + cy;
        const float pw = expf(dw) * w, ph = expf(dh) * h;
        float pb0 = pcx - 0.5f * pw, pb1 = pcy - 0.5f * ph;
        float pb2 = pcx + 0.5f * pw, pb3 = pcy + 0.5f * ph;
        float s = 0.0f;
        float d0 = pb0 - gt[di + 0], d1 = pb1 - gt[di + 1];
        float d2 = pb2 - gt[di + 2], d3 = pb3 - gt[di + 3];
        float ad;
        ad = fabsf(d0); s += (ad < 1.0f) ? 0.5f * d0 * d0 : ad - 0.5f;
        ad = fabsf(d1); s += (ad < 1.0f) ? 0.5f * d1 * d1 : ad - 0.5f;
        ad = fabsf(d2); s += (ad < 1.0f) ? 0.5f * d2 * d2 : ad - 0.5f;
        ad = fabsf(d3); s += (ad < 1.0f) ? 0.5f * d3 * d3 : ad - 0.5f;
        atomicAdd(&acc[0], s);
    }
}

// ---------------------------------------------------------------------------
// Pass B: exact k-th-largest 64-bit key per image via MSD radix select
// (8 rounds x 256-bin LDS histogram). One block per image.
// ---------------------------------------------------------------------------
__global__ __launch_bounds__(256)
void select_k(const unsigned long long* __restrict__ keys,
              const int* __restrict__ num_pos,
              unsigned long long* __restrict__ T,
              int* __restrict__ ksel)
{
    const int n   = blockIdx.x;
    const int tid = threadIdx.x;

    int k = 3 * num_pos[n];
    if (k > A_ANCH) k = A_ANCH;
    if (k <= 0) {                       // no positives -> no negatives
        if (tid == 0) { ksel[n] = 0; T[n] = ~0ull; }
        return;
    }

    __shared__ unsigned int hist[256];
    __shared__ unsigned long long sh_prefix;
    __shared__ unsigned int sh_want;

    unsigned long long prefix = 0ull;
    unsigned int want = (unsigned int)k;          // 1-based rank from the top
    const unsigned long long* krow = keys + (size_t)n * A_ANCH;

    for (int d = 7; d >= 0; --d) {
        for (int i = tid; i < 256; i += 256) hist[i] = 0u;
        __syncthreads();

        const unsigned long long himask =
            (d == 7) ? 0ull : (~0ull << (unsigned)((d + 1) * 8));
        for (int a = tid; a < A_ANCH; a += 256) {
            unsigned long long key = krow[a];
            if ((key & himask) == (prefix & himask)) {
                unsigned int b = (unsigned int)(key >> (unsigned)(d * 8)) & 0xFFu;
                atomicAdd(&hist[b], 1u);
            }
        }
        __syncthreads();

        if (tid == 0) {
            unsigned int cum = 0u; int chosen = 0;
            for (int b = 255; b >= 0; --b) {
                if (cum + hist[b] >= want) { chosen = b; break; }
                cum += hist[b];
            }
            want  -= cum;
            prefix |= ((unsigned long long)chosen) << (unsigned)(d * 8);
            sh_prefix = prefix;
            sh_want   = want;
        }
        __syncthreads();
        prefix = sh_prefix;
        want   = sh_want;
        __syncthreads();
    }

    if (tid == 0) { T[n] = prefix; ksel[n] = k; }
}

// ---------------------------------------------------------------------------
// Pass C: classification loss = sum of CE where pos || key >= threshold.
// ---------------------------------------------------------------------------
__global__ __launch_bounds__(256)
void pass_c(const float* __restrict__ ce,
            const unsigned long long* __restrict__ keys,
            const int* __restrict__ labels,
            const unsigned long long* __restrict__ T,
            const int* __restrict__ ksel,
            float* __restrict__ acc)
{
    const int n = blockIdx.y;
    const int a = blockIdx.x * 256 + threadIdx.x;
    const size_t idx = (size_t)n * A_ANCH + a;

    const bool pos = labels[idx] > 0;
    const bool neg = (ksel[n] > 0) && (keys[idx] >= T[n]);
    float v = (pos || neg) ? ce[idx] : 0.0f;

    __shared__ float red[256];
    red[threadIdx.x] = v;
    __syncthreads();
    for (int s = 128; s > 0; s >>= 1) {
        if (threadIdx.x < s) red[threadIdx.x] += red[threadIdx.x + s];
        __syncthreads();
    }
    if (threadIdx.x == 0) atomicAdd(&acc[1], red[0]);
}

// ---------------------------------------------------------------------------
// Finalize: divide by total positives, write the two outputs.
// ---------------------------------------------------------------------------
__global__ void finalize(const int* __restrict__ num_pos,
                         const float* __restrict__ acc,
                         float* __restrict__ out)
{
    if (threadIdx.x == 0 && blockIdx.x == 0) {
        int np = 0;
        for (int i = 0; i < N_IMG; ++i) np += num_pos[i];
        const float inv = 1.0f / (float)np;
        out[0] = acc[1] * inv;   // classification_loss / n_pos
        out[1] = acc[0] * inv;   // smooth_l1 / n_pos
    }
}

extern "C" void kernel_launch(void* const* d_in, const int* in_sizes, int n_in,
                              void* d_out, int out_size, void* d_ws, size_t ws_size,
                              hipStream_t stream) {
    const float* conf    = (const float*)d_in[0];
    const float* deltas  = (const float*)d_in[1];
    const float* gt      = (const float*)d_in[2];
    const float* anchors = (const float*)d_in[3];
    const int*   labels  = (const int*)d_in[4];
    float* out = (float*)d_out;

    const size_t NA = (size_t)N_IMG * A_ANCH;
    // ws layout: [ce: NA f32][keys: NA u64][small: counters]
    float*              ce    = (float*)d_ws;
    unsigned long long* keys  = (unsigned long long*)((char*)d_ws + NA * 4);
    char*               small = (char*)d_ws + NA * 12;
    int*                num_pos = (int*)small;                          // 128 B
    int*                ksel    = (int*)(small + 128);                  // 128 B
    float*              acc     = (float*)(small + 256);                //   8 B
    unsigned long long* T       = (unsigned long long*)(small + 264);   // 256 B

    // zero the accumulators/counters (capture-safe)
    hipMemsetAsync(small, 0, 264, stream);

    dim3 g1(A_ANCH / TILE, N_IMG);
    pass_a<<<g1, TILE, 0, stream>>>(conf, deltas, gt, anchors, labels,
                                    ce, keys, num_pos, acc);

    select_k<<<N_IMG, 256, 0, stream>>>(keys, num_pos, T, ksel);

    dim3 g3(A_ANCH / 256, N_IMG);
    pass_c<<<g3, 256, 0, stream>>>(ce, keys, labels, T, ksel, acc);

    finalize<<<1, 32, 0, stream>>>(num_pos, acc, out);
}